// Score_23003844837982
// MI455X (gfx1250) — compile-verified
//
#include <hip/hip_runtime.h>
#include <hip/hip_bf16.h>
#include <stdint.h>

// Shapes from the reference
#define HIDDEN 128
#define NNEI   32
#define ROW_DW 132   // 128 dwords per LDS row + 4 dwords pad (bank-conflict free)

typedef __attribute__((ext_vector_type(2))) float v2f;
typedef __attribute__((ext_vector_type(4))) float v4f;
typedef __attribute__((ext_vector_type(8))) float v8f;

// Generic pointer to LDS byte offset: flat LDS aperture keeps the allocation
// offset in addr[31:0] (ISA 10.2 aperture calc), so truncation is the offset.
__device__ __forceinline__ uint32_t lds_offset(const void* p) {
    return (uint32_t)(uintptr_t)p;
}

// One block (64 threads = 2 wave32) per node.
//   wave 0 -> neighbor rows  0..15, wave 1 -> rows 16..31
__global__ __launch_bounds__(64)
void score_softmax_kernel(const float* __restrict__ neib,   // [N, 32, 128]
                          const float* __restrict__ subg,   // [N, 128]
                          const float* __restrict__ mask,   // [N, 32]
                          const float* __restrict__ W,      // [256] (Wn | Ws)
                          const float* __restrict__ bias,   // [1]
                          float* __restrict__ out)          // [N, 32]
{
    const int node = blockIdx.x;
    const int tid  = threadIdx.x;      // 0..63
    const int lane = tid & 31;
    const int wave = tid >> 5;         // 0..1

    __shared__ float sA[NNEI * ROW_DW];   // staged neighbor tile, padded rows
    __shared__ float sWn[HIDDEN];         // first half of W
    __shared__ float sScore[NNEI];        // 32 linear scores

    // ---- Wn -> LDS (128 floats, 64 threads x 2) ----
    sWn[tid]      = W[tid];
    sWn[tid + 64] = W[tid + 64];

    // ---- stage neighbor tile via CDNA5 async DMA: global -> LDS directly ----
    // 32 rows x 128 f32; each lane copies 16B, a wave-instruction moves one
    // full 512B row. No destination VGPRs, tracked by ASYNCcnt.
    {
        const float* gN = neib + (size_t)node * (NNEI * HIDDEN);
        #pragma unroll
        for (int i = tid; i < NNEI * (HIDDEN / 4); i += 64) {
            const int row = i >> 5;            // 32 float4 per row
            const int c4  = i & 31;
            const float* src = gN + row * HIDDEN + c4 * 4;
            uint32_t dst = lds_offset(&sA[row * ROW_DW + c4 * 4]); // 16B aligned
            asm volatile("global_load_async_to_lds_b128 %0, %1, off"
                         :: "v"(dst), "v"(src)
                         : "memory");
        }
    }

    // ---- overlap with the DMA: dot(sub_graph, Ws) + b (per wave, LOADcnt) ----
    float sub;
    {
        const v4f* gs = (const v4f*)(subg + (size_t)node * HIDDEN);
        const v4f* gw = (const v4f*)(W + HIDDEN);          // Ws
        v4f a = gs[lane];
        v4f w = gw[lane];
        sub = a.x * w.x + a.y * w.y + a.z * w.z + a.w * w.w;
        #pragma unroll
        for (int off = 16; off > 0; off >>= 1)
            sub += __shfl_xor(sub, off, 32);
        sub += bias[0];
    }

    // ---- wait for async tile, then make it visible to both waves ----
#if __has_builtin(__builtin_amdgcn_s_wait_asynccnt)
    __builtin_amdgcn_s_wait_asynccnt(0);
#else
    asm volatile("s_wait_asynccnt 0" ::: "memory");
#endif
    __syncthreads();

    // ---- WMMA chain: D += A(16x4 f32) x B(4x16 f32), K = 0..128 step 4 ----
    // A layout (ISA 7.12.2): lane L<16 holds row M=L, K={k,k+1}; lane L>=16
    // holds row M=L-16, K={k+2,k+3}. B mirrors A's per-lane K mapping; all 16
    // columns of B carry the same Wn values, so every accumulator column ends
    // up holding dot(row, Wn).
    const int   rbase = wave * 16;
    const int   mrow  = lane & 15;
    const int   khalf = (lane >> 4) * 2;        // 0 or 2
    const float* arow = &sA[(rbase + mrow) * ROW_DW];

    v8f acc = {0.f, 0.f, 0.f, 0.f, 0.f, 0.f, 0.f, 0.f};
    #pragma unroll
    for (int k = 0; k < HIDDEN; k += 4) {
        v2f a = *(const v2f*)&arow[k + khalf];  // ds_load_b64, conflict-free
        v2f b = *(const v2f*)&sWn[k + khalf];   // broadcast within half-wave
        acc = __builtin_amdgcn_wmma_f32_16x16x4_f32(
                  /*neg_a=*/false, a, /*neg_b=*/false, b,
                  /*c_mod=*/(short)0, acc,
                  /*reuse_a=*/false, /*reuse_b=*/false);
    }

    // ---- extract column N=0: lane 0 holds rows 0..7, lane 16 rows 8..15 ----
    if ((lane & 15) == 0) {
        const int roff = rbase + (lane >> 4) * 8;
        #pragma unroll
        for (int i = 0; i < 8; ++i)
            sScore[roff + i] = acc[i] + sub;    // linear score (pre-sigmoid)
    }
    __syncthreads();

    // ---- sigmoid + mask + softmax over the 32 neighbors (wave 0 only) ----
    if (wave == 0) {
        float s = sScore[lane];
        s = 1.f / (1.f + expf(-s));                       // sigmoid
        s += mask[(size_t)node * NNEI + lane];            // pad mask (zeros)

        float m = s;
        #pragma unroll
        for (int off = 16; off > 0; off >>= 1)
            m = fmaxf(m, __shfl_xor(m, off, 32));
        float e = expf(s - m);
        float t = e;
        #pragma unroll
        for (int off = 16; off > 0; off >>= 1)
            t += __shfl_xor(t, off, 32);

        out[(size_t)node * NNEI + lane] = e / t;
    }
}

extern "C" void kernel_launch(void* const* d_in, const int* in_sizes, int n_in,
                              void* d_out, int out_size, void* d_ws, size_t ws_size,
                              hipStream_t stream) {
    // setup_inputs order: x(0), neibour(1), sub_graph(2), pmi(3),
    //                     nei_pad_mask(4), W(5), b(6)
    const float* neib = (const float*)d_in[1];
    const float* subg = (const float*)d_in[2];
    const float* mask = (const float*)d_in[4];
    const float* W    = (const float*)d_in[5];
    const float* bias = (const float*)d_in[6];
    float*       out  = (float*)d_out;

    const int n_nodes = in_sizes[2] / HIDDEN;   // sub_graph has N*128 elements

    score_softmax_kernel<<<n_nodes, 64, 0, stream>>>(neib, subg, mask, W, bias, out);
}